// BatchNormalizationLayer_87076166959941
// MI455X (gfx1250) — compile-verified
//
#include <hip/hip_runtime.h>

// Problem constants (from reference): X is [N_ROWS, N_COLS] float32, row-major.
#define N_ROWS 131072
#define N_COLS 512
#define SLAB   1024   // rows reduced per block in pass 1

typedef __attribute__((ext_vector_type(2))) float v2f;
typedef __attribute__((ext_vector_type(4))) float f4;
typedef __attribute__((ext_vector_type(8))) float v8f;

// ---------------------------------------------------------------------------
// Pass 0: zero the accumulator workspace (sum[512] + sumsq[512]).
// ---------------------------------------------------------------------------
__global__ void bn_zero_ws_kernel(float* __restrict__ ws) {
    ws[blockIdx.x * blockDim.x + threadIdx.x] = 0.0f;
}

// ---------------------------------------------------------------------------
// Pass 1: per-column sum and sum-of-squares via V_WMMA_F32_16X16X4_F32.
//   D = A(16x4 ones) x B(4x16 tile of X) + C  => every row of D accumulates
//   the column sums of the tile. Two accumulator chains: raw values and
//   squares. Each wave owns 16 consecutive columns and a 1024-row slab.
// Layout (wave32):
//   A 16x4 f32: 2 VGPRs, lane m holds row m; VGPR0={K0|K2}, VGPR1={K1|K3}.
//   B  4x16 f32: 2 VGPRs, mirrored: lanes 0-15 rows K0/K1, lanes 16-31 K2/K3,
//                column N = lane%16.
//   C/D 16x16 f32: VGPR0 lanes 0-15 = (M=0, N=lane) -> column sums live in
//                  acc[0] of lanes 0..15.
// ---------------------------------------------------------------------------
__global__ __launch_bounds__(256) void bn_colstat_wmma_kernel(
        const float* __restrict__ X,
        float* __restrict__ wsSum,
        float* __restrict__ wsSq) {
    const int tid  = threadIdx.x;
    const int wave = tid >> 5;
    const int lane = tid & 31;

    const int colGroup = blockIdx.y * 8 + wave;        // 0..31 (32 groups of 16 cols)
    const int col      = colGroup * 16 + (lane & 15);  // 0..511
    const int k0       = (lane >> 4) * 2;              // 0 (lanes 0-15) or 2 (lanes 16-31)
    const size_t rowBase = (size_t)blockIdx.x * SLAB;

    const float* p0 = X + (rowBase + (size_t)k0) * N_COLS + col; // row k0 of tile
    const float* p1 = p0 + N_COLS;                               // row k0+1 of tile

    v2f a; a[0] = 1.0f; a[1] = 1.0f;   // A = ones(16x4)
    v8f accS = {};                     // column sums
    v8f accQ = {};                     // column sums of squares

    for (int r = 0; r < SLAB; r += 4) {
        // 2KB row stride defeats naive prefetch: hint 32 rows ahead
        // (emits global_prefetch_b8).
        __builtin_prefetch(p0 + 32 * N_COLS, 0, 1);

        v2f b;
        b[0] = p0[0];
        b[1] = p1[0];
        v2f q = b * b;

        accS = __builtin_amdgcn_wmma_f32_16x16x4_f32(
                   false, a, false, b, (short)0, accS, false, false);
        accQ = __builtin_amdgcn_wmma_f32_16x16x4_f32(
                   false, a, false, q, (short)0, accQ, false, false);

        p0 += 4 * N_COLS;
        p1 += 4 * N_COLS;
    }

    // D row M=0 lives in acc[0] of lanes 0..15 (N = lane). All D rows are
    // identical (A is all-ones), so lanes 0..15 suffice.
    if (lane < 16) {
        atomicAdd(&wsSum[col], accS[0]);
        atomicAdd(&wsSq[col],  accQ[0]);
    }
}

// ---------------------------------------------------------------------------
// Pass 2: fold sums into per-column scale/shift:
//   scale = gamma * rsqrt(E[x^2] - mean^2),  shift = beta - mean*scale
// ---------------------------------------------------------------------------
__global__ void bn_finalize_kernel(const float* __restrict__ wsSum,
                                   const float* __restrict__ wsSq,
                                   const float* __restrict__ gamma,
                                   const float* __restrict__ beta,
                                   float* __restrict__ coef) {
    const int d = threadIdx.x;                 // blockDim.x == 512
    const float invN = 1.0f / (float)N_ROWS;
    const float mean = wsSum[d] * invN;
    const float var  = fmaf(-mean, mean, wsSq[d] * invN);
    const float s    = gamma[d] * rsqrtf(var); // reference has no epsilon
    coef[d]          = s;
    coef[N_COLS + d] = fmaf(-mean, s, beta[d]);
}

// ---------------------------------------------------------------------------
// Pass 3: Y = X*scale[col] + shift[col].  128-bit vector loads/stores,
// non-temporal (X already consumed once; Y never re-read) so L2 stays
// available for the coefficient broadcast.
// ---------------------------------------------------------------------------
__global__ __launch_bounds__(256) void bn_normalize_kernel(
        const f4* __restrict__ X4,
        const float* __restrict__ coef,
        f4* __restrict__ Y4) {
    const unsigned i = blockIdx.x * 256u + threadIdx.x;  // float4 index
    const unsigned c = i & 127u;                         // (i*4) % 512 / 4

    const f4* coefS = (const f4*)coef;         // scale[512]
    const f4* coefT = coefS + (N_COLS / 4);    // shift[512]

    f4 x = __builtin_nontemporal_load(X4 + i);
    f4 s = coefS[c];
    f4 t = coefT[c];
    f4 y = x * s + t;                          // contracts to v_fma_f32
    __builtin_nontemporal_store(y, Y4 + i);
}

// ---------------------------------------------------------------------------
// Host-side launcher. d_in = {X[N*D], gamma[D], beta[D]} f32; d_out = Y f32.
// Workspace layout (floats): sum[512] | sumsq[512] | scale[512] | shift[512]
// ---------------------------------------------------------------------------
extern "C" void kernel_launch(void* const* d_in, const int* in_sizes, int n_in,
                              void* d_out, int out_size, void* d_ws, size_t ws_size,
                              hipStream_t stream) {
    const float* X     = (const float*)d_in[0];
    const float* gamma = (const float*)d_in[1];
    const float* beta  = (const float*)d_in[2];
    float* Y  = (float*)d_out;
    float* ws = (float*)d_ws;

    float* wsSum = ws;                 // [512]
    float* wsSq  = ws + N_COLS;        // [512]
    float* coef  = ws + 2 * N_COLS;    // [1024] = scale | shift

    // Pass 0: zero accumulators (1024 floats).
    bn_zero_ws_kernel<<<4, 256, 0, stream>>>(ws);

    // Pass 1: WMMA column reduction. 128 row-slabs x 4 column-quadrants,
    // 8 waves/block (one 16-column group per wave).
    dim3 g1(N_ROWS / SLAB, 4, 1);
    bn_colstat_wmma_kernel<<<g1, 256, 0, stream>>>(X, wsSum, wsSq);

    // Pass 2: 512 columns -> scale/shift.
    bn_finalize_kernel<<<1, N_COLS, 0, stream>>>(wsSum, wsSq, gamma, beta, coef);

    // Pass 3: streaming normalize, one float4 per thread.
    const unsigned nvec4 = (unsigned)N_ROWS * (unsigned)N_COLS / 4u; // 16.7M
    bn_normalize_kernel<<<nvec4 / 256u, 256, 0, stream>>>((const f4*)X, coef, (f4*)Y);
}